// TDNNGeneratorQAT_58540404245068
// MI455X (gfx1250) — compile-verified
//
#include <hip/hip_runtime.h>
#include <math.h>

typedef __attribute__((ext_vector_type(2))) float v2f;
typedef __attribute__((ext_vector_type(8))) float v8f;

#define QMAXF 32767.0f
#define QMINF (-32768.0f)
#define EPSQ  1e-8f
#define LEAKYF 0.2f

// ws layout (floats): [0..3] absmax slots (x,h1,h2,h3); quantized weights follow.
// Cached int16 x-codes (optional) live at byte offset 4096.
#define WS_W1Q 8            // 32x30
#define WS_W2Q (8+960)      // 16x32
#define WS_W3Q (8+960+512)  // 2x16
#define WS_XQ_BYTE_OFF 4096

__device__ __forceinline__ float q_scale(float absmax) {
    return fmaxf(absmax, EPSQ) / QMAXF;
}
// quant code (integer-valued float in [-32768,32767])
__device__ __forceinline__ float fqr(float v, float inv) {
    float r = rintf(v * inv);
    return __builtin_amdgcn_fmed3f(r, QMINF, QMAXF);
}
// full fake-quantize with hoisted reciprocal
__device__ __forceinline__ float fqi(float v, float inv, float s) {
    return fqr(v, inv) * s;
}
// exact divide version (tiny, cold paths: weight quant)
__device__ __forceinline__ float fq_div(float v, float s) {
    float r = rintf(v / s);
    return __builtin_amdgcn_fmed3f(r, QMINF, QMAXF) * s;
}
__device__ __forceinline__ float leaky(float v) { return fmaxf(v, LEAKYF * v); }

// ---------------- per-channel weight quantization + absmax-slot reset ----------------
__global__ void quant_weights_kernel(const float* __restrict__ W1,
                                     const float* __restrict__ W2,
                                     const float* __restrict__ W3,
                                     float* __restrict__ ws) {
    int t = threadIdx.x;
    if (t < 4) ws[t] = 0.0f;                 // reset absmax accumulators each call
    if (t < 32) {                            // W1: 32 channels x 30
        const float* w = W1 + t * 30; float* d = ws + WS_W1Q + t * 30;
        float m = 0.f;
        for (int k = 0; k < 30; ++k) m = fmaxf(m, fabsf(w[k]));
        float s = q_scale(m);
        for (int k = 0; k < 30; ++k) d[k] = fq_div(w[k], s);
    } else if (t < 48) {                     // W2: 16 channels x 32
        int c = t - 32;
        const float* w = W2 + c * 32; float* d = ws + WS_W2Q + c * 32;
        float m = 0.f;
        for (int k = 0; k < 32; ++k) m = fmaxf(m, fabsf(w[k]));
        float s = q_scale(m);
        for (int k = 0; k < 32; ++k) d[k] = fq_div(w[k], s);
    } else if (t < 50) {                     // W3: 2 channels x 16
        int c = t - 48;
        const float* w = W3 + c * 16; float* d = ws + WS_W3Q + c * 16;
        float m = 0.f;
        for (int k = 0; k < 16; ++k) m = fmaxf(m, fabsf(w[k]));
        float s = q_scale(m);
        for (int k = 0; k < 16; ++k) d[k] = fq_div(w[k], s);
    }
}

// ---------------- global absmax over x ----------------
__global__ __launch_bounds__(256)
void absmax_kernel(const float* __restrict__ x, long long n, unsigned int* __restrict__ amax) {
    float m = 0.f;
    long long i = (long long)blockIdx.x * blockDim.x + threadIdx.x;
    long long stride = (long long)gridDim.x * blockDim.x;
    long long n4 = n >> 2;
    const float4* x4 = (const float4*)x;
    for (long long j = i; j < n4; j += stride) {
        float4 v = x4[j];
        m = fmaxf(m, fmaxf(fmaxf(fabsf(v.x), fabsf(v.y)), fmaxf(fabsf(v.z), fabsf(v.w))));
    }
    for (long long j = (n4 << 2) + i; j < n; j += stride) m = fmaxf(m, fabsf(x[j]));
    for (int off = 16; off > 0; off >>= 1) m = fmaxf(m, __shfl_xor(m, off, 32));
    __shared__ float smax[8];
    int lane = threadIdx.x & 31, wid = threadIdx.x >> 5;
    if (lane == 0) smax[wid] = m;
    __syncthreads();
    if (wid == 0) {
        m = (lane < (int)(blockDim.x >> 5)) ? smax[lane] : 0.f;
        for (int off = 16; off > 0; off >>= 1) m = fmaxf(m, __shfl_xor(m, off, 32));
        if (lane == 0) atomicMax(amax, __float_as_uint(m));
    }
}

// ---------------- fused pipeline; one wave per block; WMMA f32 16x16x4 ----------------
// STAGE 1: absmax(h1) -> ws[1] (and, if CACHEX, store int16 x-codes)
// STAGE 2: absmax(h2) -> ws[2];  STAGE 3: absmax(h3) -> ws[3];  STAGE 4: write output.
template <int STAGE, bool CACHEX>
__global__ __launch_bounds__(32)
void pipeline(const float* __restrict__ x, short* __restrict__ xq,
              const float* __restrict__ b1, const float* __restrict__ b2,
              const float* __restrict__ b3,
              float* __restrict__ out, float* __restrict__ ws, int ntiles) {
    const float* W1q = ws + WS_W1Q;
    const float* W2q = ws + WS_W2Q;
    const float* W3q = ws + WS_W3Q;

    __shared__ float lds1[16 * 34];  // h1 tile (16 x 32), padded stride
    __shared__ float lds2[16 * 18];  // h2 tile (16 x 16), padded stride
    __shared__ float ldso[32];       // out tile (16 x 2)

    const int lane = threadIdx.x;
    const int m    = lane & 15;      // M for A-frag / N for B-frag / N for C,D
    const int hi   = lane >> 4;      // lane-half
    const int koff = hi * 2;         // K offset within a 4-wide step

    const float sx = q_scale(ws[0]);
    const float invx = 1.0f / sx;
    float s1 = 0.f, inv1 = 0.f, s2 = 0.f, inv2 = 0.f;
    if (STAGE >= 2) { s1 = q_scale(ws[1]); inv1 = 1.0f / s1; }
    if (STAGE >= 3) { s2 = q_scale(ws[2]); inv2 = 1.0f / s2; }
    float s3 = 0.f, inv3 = 0.f, so = 0.f, invo = 0.f;
    if (STAGE >= 4) {
        s3 = q_scale(ws[3]); inv3 = 1.0f / s3;
        // fake-quant preserves absmax exactly as 32767*s3; tanh is odd & monotone
        so = q_scale(tanhf(QMAXF * s3)); invo = 1.0f / so;
    }

    // Preload B fragments (quantized weights). B[k][n]: lane half gives K = kb + j + koff.
    v2f b1r[8][2];
    for (int ks = 0; ks < 8; ++ks) {
        int kb = ks * 4 + koff;
        for (int t = 0; t < 2; ++t) {
            int n = m + 16 * t;
            if (kb < 30) { b1r[ks][t].x = W1q[n * 30 + kb]; b1r[ks][t].y = W1q[n * 30 + kb + 1]; }
            else         { b1r[ks][t].x = 0.f;              b1r[ks][t].y = 0.f; }
        }
    }
    v2f b2r[8];
    for (int ks = 0; ks < 8; ++ks) {
        int kb = ks * 4 + koff;
        b2r[ks].x = W2q[m * 32 + kb]; b2r[ks].y = W2q[m * 32 + kb + 1];
    }
    v2f b3r[4];
    for (int ks = 0; ks < 4; ++ks) {
        int kb = ks * 4 + koff;
        if (m < 2) { b3r[ks].x = W3q[m * 16 + kb]; b3r[ks].y = W3q[m * 16 + kb + 1]; }
        else       { b3r[ks].x = 0.f;              b3r[ks].y = 0.f; }
    }
    const float bias1a = b1[m], bias1b = b1[m + 16];
    const float bias2  = b2[m];
    const float bias3  = (m < 2) ? b3[m] : 0.f;

    float localmax = 0.f;

    for (int tile = blockIdx.x; tile < ntiles; tile += gridDim.x) {
        const long long row0 = (long long)tile * 16;
        const float* xr  = x + (row0 + m) * 30;
        short*       xqw = CACHEX ? (xq + (row0 + m) * 30) : (short*)0;

        // ---- layer 1: (16x30)x(30x32) via 8 K-steps, two N-tiles ----
        v8f acc0 = {0,0,0,0,0,0,0,0}, acc1 = {0,0,0,0,0,0,0,0};
        for (int ks = 0; ks < 8; ++ks) {
            int kb = ks * 4 + koff;
            v2f a;
            if (kb < 30) {
                if (CACHEX && STAGE >= 2) {
                    short2 p = *(const short2*)(xqw + kb);      // cached quant codes
                    a.x = (float)p.x * sx; a.y = (float)p.y * sx;
                } else {
                    float r0 = fqr(xr[kb], invx), r1 = fqr(xr[kb + 1], invx);
                    a.x = r0 * sx; a.y = r1 * sx;
                    if (CACHEX && STAGE == 1) {
                        short2 p; p.x = (short)r0; p.y = (short)r1;
                        *(short2*)(xqw + kb) = p;
                    }
                }
            } else { a.x = 0.f; a.y = 0.f; }
            acc0 = __builtin_amdgcn_wmma_f32_16x16x4_f32(false, a, false, b1r[ks][0],
                                                         (short)0, acc0, false, false);
            acc1 = __builtin_amdgcn_wmma_f32_16x16x4_f32(false, a, false, b1r[ks][1],
                                                         (short)0, acc1, false, false);
        }
        for (int i = 0; i < 8; ++i) { acc0[i] += bias1a; acc1[i] += bias1b; }

        if (STAGE == 1) {
            for (int i = 0; i < 8; ++i)
                localmax = fmaxf(localmax, fmaxf(fabsf(acc0[i]), fabsf(acc1[i])));
            continue;
        }

        // quantize + LeakyReLU, transpose D-layout -> row-major tile in LDS
        for (int i = 0; i < 8; ++i) {
            float v0 = leaky(fqi(acc0[i], inv1, s1));
            float v1 = leaky(fqi(acc1[i], inv1, s1));
            int rr = i + hi * 8;
            lds1[rr * 34 + m]      = v0;
            lds1[rr * 34 + m + 16] = v1;
        }
        __syncthreads();

        // ---- layer 2: (16x32)x(32x16) ----
        v8f acc2 = {0,0,0,0,0,0,0,0};
        for (int ks = 0; ks < 8; ++ks) {
            int kb = ks * 4 + koff;
            v2f a; a.x = lds1[m * 34 + kb]; a.y = lds1[m * 34 + kb + 1];
            acc2 = __builtin_amdgcn_wmma_f32_16x16x4_f32(false, a, false, b2r[ks],
                                                         (short)0, acc2, false, false);
        }
        for (int i = 0; i < 8; ++i) acc2[i] += bias2;

        if (STAGE == 2) {
            for (int i = 0; i < 8; ++i) localmax = fmaxf(localmax, fabsf(acc2[i]));
            __syncthreads();
            continue;
        }

        for (int i = 0; i < 8; ++i) {
            float v = leaky(fqi(acc2[i], inv2, s2));
            lds2[(i + hi * 8) * 18 + m] = v;
        }
        __syncthreads();

        // ---- layer 3: (16x16)x(16x2), B columns >=2 are zero ----
        v8f acc3 = {0,0,0,0,0,0,0,0};
        for (int ks = 0; ks < 4; ++ks) {
            int kb = ks * 4 + koff;
            v2f a; a.x = lds2[m * 18 + kb]; a.y = lds2[m * 18 + kb + 1];
            acc3 = __builtin_amdgcn_wmma_f32_16x16x4_f32(false, a, false, b3r[ks],
                                                         (short)0, acc3, false, false);
        }
        for (int i = 0; i < 8; ++i) acc3[i] += bias3;

        if (STAGE == 3) {
            for (int i = 0; i < 8; ++i) localmax = fmaxf(localmax, fabsf(acc3[i]));
            __syncthreads();
            continue;
        }

        // ---- STAGE 4: quantize -> tanh -> output quantize -> coalesced store ----
        for (int i = 0; i < 8; ++i) {
            float v = fqi(acc3[i], inv3, s3);
            v = tanhf(v);
            v = fqi(v, invo, so);
            if (m < 2) ldso[(i + hi * 8) * 2 + m] = v;
        }
        __syncthreads();
        out[row0 * 2 + lane] = ldso[lane];   // 128B contiguous per wave
        __syncthreads();
    }

    if (STAGE >= 1 && STAGE <= 3) {
        for (int off = 16; off > 0; off >>= 1)
            localmax = fmaxf(localmax, __shfl_xor(localmax, off, 32));
        if (lane == 0)
            atomicMax((unsigned int*)(ws + STAGE), __float_as_uint(localmax));
    }
}

extern "C" void kernel_launch(void* const* d_in, const int* in_sizes, int n_in,
                              void* d_out, int out_size, void* d_ws, size_t ws_size,
                              hipStream_t stream) {
    const float* x  = (const float*)d_in[0];
    const float* W1 = (const float*)d_in[1];
    const float* b1 = (const float*)d_in[2];
    const float* W2 = (const float*)d_in[3];
    const float* b2 = (const float*)d_in[4];
    const float* W3 = (const float*)d_in[5];
    const float* b3 = (const float*)d_in[6];
    float* out = (float*)d_out;
    float* ws  = (float*)d_ws;

    const long long nx = (long long)in_sizes[0];
    const long long nrows = nx / 30;
    const int ntiles = (int)(nrows / 16);      // 4M rows -> 250,000 full tiles

    quant_weights_kernel<<<1, 64, 0, stream>>>(W1, W2, W3, ws);
    absmax_kernel<<<2048, 256, 0, stream>>>(x, nx, (unsigned int*)ws);

    const int blocks = (ntiles > 0) ? (ntiles < 4096 ? ntiles : 4096) : 1;
    const bool cachex =
        ws_size >= (size_t)WS_XQ_BYTE_OFF + (size_t)nrows * 30 * sizeof(short);
    short* xq = (short*)((char*)d_ws + WS_XQ_BYTE_OFF);

    if (cachex) {
        pipeline<1, true><<<blocks, 32, 0, stream>>>(x, xq, b1, b2, b3, out, ws, ntiles);
        pipeline<2, true><<<blocks, 32, 0, stream>>>(x, xq, b1, b2, b3, out, ws, ntiles);
        pipeline<3, true><<<blocks, 32, 0, stream>>>(x, xq, b1, b2, b3, out, ws, ntiles);
        pipeline<4, true><<<blocks, 32, 0, stream>>>(x, xq, b1, b2, b3, out, ws, ntiles);
    } else {
        pipeline<1, false><<<blocks, 32, 0, stream>>>(x, (short*)0, b1, b2, b3, out, ws, ntiles);
        pipeline<2, false><<<blocks, 32, 0, stream>>>(x, (short*)0, b1, b2, b3, out, ws, ntiles);
        pipeline<3, false><<<blocks, 32, 0, stream>>>(x, (short*)0, b1, b2, b3, out, ws, ntiles);
        pipeline<4, false><<<blocks, 32, 0, stream>>>(x, (short*)0, b1, b2, b3, out, ws, ntiles);
    }
}